// WeightedMSELoss_60335700574782
// MI455X (gfx1250) — compile-verified
//
#include <hip/hip_runtime.h>

// Use ext_vector types (HIP's float4 is a class; the clang builtins want these).
typedef __attribute__((ext_vector_type(4))) float f4;
typedef __attribute__((ext_vector_type(2))) float v2f;
typedef __attribute__((ext_vector_type(8))) float v8f;

// ---------------------------------------------------------------------------
// Wave-wide sum via V_WMMA_F32_16X16X4_F32 (CDNA5 matrix path).
//
// A-matrix (16x4 f32, 2 VGPRs): lane L<16 holds A[L][0],A[L][1]; lane L>=16
// holds A[L-16][2],A[L-16][3].  Put the per-lane partial in a[0], 0 in a[1],
// and B = all-ones => D[m][n] = sum_k A[m][k] = p_m + p_{m+16}.
// D layout: lane n (<16), VGPR v holds D[v][n]; lane n+16, VGPR v holds
// D[v+8][n].  So s = sum(d[0..7]) gives lanes 0-15: sum_{m=0..7}(p_m+p_{m+16})
// and lanes 16-31: sum_{m=8..15}(p_m+p_{m+16}); s + shfl_xor(s,16) = wave sum
// in every lane.  EXEC must be all 1s at the call site (it is: uniform control
// flow / reconverged loops).
// ---------------------------------------------------------------------------
__device__ __forceinline__ float wave_sum_wmma(float p) {
  v2f a; a.x = p;    a.y = 0.0f;
  v2f b; b.x = 1.0f; b.y = 1.0f;
  v8f c = {};
  // 8 args: (neg_a, A, neg_b, B, c_mod, C, reuse_a, reuse_b)
  v8f d = __builtin_amdgcn_wmma_f32_16x16x4_f32(
      /*neg_a=*/false, a, /*neg_b=*/false, b,
      /*c_mod=*/(short)0, c, /*reuse_a=*/false, /*reuse_b=*/false);
  float s = ((d[0] + d[1]) + (d[2] + d[3])) + ((d[4] + d[5]) + (d[6] + d[7]));
  s += __shfl_xor(s, 16, 32);
  return s;  // full 32-lane sum, broadcast to all lanes
}

// Branchless bin weight: for t in (e_i, e_{i+1}] -> w_i ; outside (e0, e5] -> 0.
__device__ __forceinline__ float pick_w(float t,
                                        float e0, float e1, float e2,
                                        float e3, float e4, float e5,
                                        float w0, float w1, float w2,
                                        float w3, float w4) {
  float w = 0.0f;
  w = (t > e0) ? w0 : w;
  w = (t > e1) ? w1 : w;
  w = (t > e2) ? w2 : w;
  w = (t > e3) ? w3 : w;
  w = (t > e4) ? w4 : w;
  w = (t > e5) ? 0.0f : w;
  return w;
}

__global__ __launch_bounds__(256) void wmse_partial_kernel(
    const f4* __restrict__ pred4, const f4* __restrict__ targ4,
    const float* __restrict__ weights, const float* __restrict__ edge,
    float* __restrict__ partials, int n4, int nTail, const float* __restrict__ predS,
    const float* __restrict__ targS) {
  // Uniform-address loads of the tiny tables (cached / scalarized).
  const float e0 = edge[0], e1 = edge[1], e2 = edge[2];
  const float e3 = edge[3], e4 = edge[4], e5 = edge[5];
  const float w0 = weights[0], w1 = weights[1], w2 = weights[2];
  const float w3 = weights[3], w4 = weights[4];

  const int stride = (int)(gridDim.x * blockDim.x);
  const int gtid   = (int)(blockIdx.x * blockDim.x + threadIdx.x);

  float acc = 0.0f;
  // Main streaming loop: 32 B per lane per iteration, non-temporal (data is
  // read-once and larger than the 192 MB L2 -> don't rinse the cache).
  for (int i = gtid; i < n4; i += stride) {
    f4 p = __builtin_nontemporal_load(pred4 + i);
    f4 t = __builtin_nontemporal_load(targ4 + i);
    float d0 = p.x - t.x, d1 = p.y - t.y, d2 = p.z - t.z, d3 = p.w - t.w;
    acc = fmaf(pick_w(t.x, e0,e1,e2,e3,e4,e5, w0,w1,w2,w3,w4) * d0, d0, acc);
    acc = fmaf(pick_w(t.y, e0,e1,e2,e3,e4,e5, w0,w1,w2,w3,w4) * d1, d1, acc);
    acc = fmaf(pick_w(t.z, e0,e1,e2,e3,e4,e5, w0,w1,w2,w3,w4) * d2, d2, acc);
    acc = fmaf(pick_w(t.w, e0,e1,e2,e3,e4,e5, w0,w1,w2,w3,w4) * d3, d3, acc);
  }
  // Scalar tail (empty for N = 2^25; kept for generality).
  for (int i = 4 * n4 + gtid; i < 4 * n4 + nTail; i += stride) {
    float p = predS[i], t = targS[i];
    float d = p - t;
    acc = fmaf(pick_w(t, e0,e1,e2,e3,e4,e5, w0,w1,w2,w3,w4) * d, d, acc);
  }

  // Loops have reconverged: EXEC is all 1s -> legal to issue WMMA.
  float s = wave_sum_wmma(acc);

  __shared__ float lsum[8];
  const int lane = (int)(threadIdx.x & 31u);
  const int wid  = (int)(threadIdx.x >> 5u);
  if (lane == 0) lsum[wid] = s;
  __syncthreads();
  if (threadIdx.x == 0) {
    float tsum = 0.0f;
#pragma unroll
    for (int i = 0; i < 8; ++i) tsum += lsum[i];
    partials[blockIdx.x] = tsum;  // one deterministic partial per block
  }
}

__global__ __launch_bounds__(256) void wmse_final_kernel(
    const float* __restrict__ partials, const float* __restrict__ weights,
    float* __restrict__ out, int nPartials, int nW) {
  float p = 0.0f;
  for (int i = (int)threadIdx.x; i < nPartials; i += 256) p += partials[i];

  float s = wave_sum_wmma(p);

  __shared__ float lsum[8];
  const int lane = (int)(threadIdx.x & 31u);
  const int wid  = (int)(threadIdx.x >> 5u);
  if (lane == 0) lsum[wid] = s;
  __syncthreads();
  if (threadIdx.x == 0) {
    float tsum = 0.0f;
#pragma unroll
    for (int i = 0; i < 8; ++i) tsum += lsum[i];
    float wsum = 0.0f;
    for (int i = 0; i < nW; ++i) wsum += weights[i];
    out[0] = tsum / wsum;
  }
}

extern "C" void kernel_launch(void* const* d_in, const int* in_sizes, int n_in,
                              void* d_out, int out_size, void* d_ws, size_t ws_size,
                              hipStream_t stream) {
  (void)n_in; (void)out_size; (void)ws_size;
  const float* pred    = (const float*)d_in[0];
  const float* targ    = (const float*)d_in[1];
  const float* weights = (const float*)d_in[2];
  const float* edge    = (const float*)d_in[3];
  (void)edge;
  float* out      = (float*)d_out;
  float* partials = (float*)d_ws;  // blocks * 4 bytes = 8 KB of scratch

  const int n     = in_sizes[0];
  const int n4    = n >> 2;        // float4 count (N = 2^25 is divisible by 4)
  const int nTail = n - 4 * n4;
  const int nW    = in_sizes[2];

  const int threads = 256;         // 8 wave32s per workgroup
  const int blocks  = 2048;        // 524288 threads -> plenty of HBM requests in flight

  wmse_partial_kernel<<<blocks, threads, 0, stream>>>(
      (const f4*)pred, (const f4*)targ, weights, edge, partials, n4, nTail,
      pred, targ);
  wmse_final_kernel<<<1, threads, 0, stream>>>(partials, weights, out, blocks, nW);
}